// SpectralClassifier_13400297964007
// MI455X (gfx1250) — compile-verified
//
#include <hip/hip_runtime.h>
#include <math.h>
#include <stdint.h>

#ifndef M_PI
#define M_PI 3.14159265358979323846
#endif

typedef __attribute__((ext_vector_type(2))) float v2f;
typedef __attribute__((ext_vector_type(8))) float v8f;

#define BSZ   1024
#define SSZ   128
#define DSZ   768
#define NLEN  127      // S-1
#define KLO   8
#define KHI   48
#define H1SZ  256
#define H2SZ  64

// ---------------------------------------------------------------------------
// Kernel 0: collapse mean∘IDCT∘bandpass∘DCT into a single length-127 weight
// vector A[m] = (1/(2N^2)) * sum_{k in [KLO,KHI)} S3[k] * C2[k][m]
// where S3[k] = sum_n C3[n][k], C2[k][n] = 2cos(pi k (2n+1)/(2N)).
// ---------------------------------------------------------------------------
__global__ void spec_weights_kernel(float* __restrict__ wA) {
  __shared__ float s3[KHI - KLO];
  const int t = threadIdx.x;
  if (t < (KHI - KLO)) {
    const int k = t + KLO;
    double acc = 0.0;
    for (int n = 0; n < NLEN; ++n)
      acc += 2.0 * cos(M_PI * (double)k * (2.0 * n + 1.0) / (2.0 * NLEN));
    s3[t] = (float)acc;
  }
  __syncthreads();
  if (t < NLEN) {
    double acc = 0.0;
    for (int j = 0; j < (KHI - KLO); ++j) {
      const int k = j + KLO;
      acc += (double)s3[j] *
             2.0 * cos(M_PI * (double)k * (2.0 * t + 1.0) / (2.0 * NLEN));
    }
    wA[t] = (float)(acc / (2.0 * (double)NLEN * (double)NLEN));
  }
}

// ---------------------------------------------------------------------------
// Kernel 1: vectors[b,d] = sum_n wA[n] * mask[b,n+1] * embeds[b,n+1,d]
// Bandwidth bound: streams 402MB of embeds once, float4-coalesced (~17us at
// 23.3 TB/s). 192 threads/block (6 wave32s) cover 768 floats as float4 lanes.
// ---------------------------------------------------------------------------
__global__ void __launch_bounds__(192)
spec_reduce_kernel(const float* __restrict__ embeds,
                   const uint8_t* __restrict__ mask,
                   const float* __restrict__ wA,
                   float* __restrict__ vectors) {
  __shared__ float wl[NLEN];
  const int b = blockIdx.x;
  const int t = threadIdx.x;
  if (t < NLEN)
    wl[t] = mask[b * SSZ + t + 1] ? wA[t] : 0.0f;
  __syncthreads();

  const float4* __restrict__ row =
      (const float4*)(embeds + ((size_t)b * SSZ + 1) * DSZ);
  const int rstride = DSZ / 4;  // 192 float4 per sequence row

  float4 acc = make_float4(0.f, 0.f, 0.f, 0.f);
  for (int n = 0; n < NLEN; ++n) {
    if (n + 4 < NLEN)
      __builtin_prefetch(&row[(n + 4) * rstride + t], 0, 0);  // global_prefetch_b8
    const float w = wl[n];
    const float4 v = row[n * rstride + t];
    acc.x = fmaf(w, v.x, acc.x);
    acc.y = fmaf(w, v.y, acc.y);
    acc.z = fmaf(w, v.z, acc.z);
    acc.w = fmaf(w, v.w, acc.w);
  }
  ((float4*)(vectors + (size_t)b * DSZ))[t] = acc;
}

// ---------------------------------------------------------------------------
// One 16x16 output tile via V_WMMA_F32_16X16X4_F32.
// A (16xK) read from LDS, B (KxN slice) read from global, column 'Bcol'.
// Per ISA layout: lanes 0-15 carry K=k0,k0+1 ; lanes 16-31 carry K=k0+2,k0+3.
// ---------------------------------------------------------------------------
__device__ __forceinline__ v8f wmma16x16_f32(const float* __restrict__ A,
                                             int lda,
                                             const float* __restrict__ Bcol,
                                             int ldb, int K, int lane) {
  v8f c = {};
  const int row  = lane & 15;
  const int koff = (lane >> 4) << 1;  // 0 for lanes 0-15, 2 for lanes 16-31
#pragma unroll 4
  for (int k0 = 0; k0 < K; k0 += 4) {
    const int ka = k0 + koff;
    v2f a, b;
    a.x = A[row * lda + ka];
    a.y = A[row * lda + ka + 1];
    b.x = Bcol[ka * ldb];
    b.y = Bcol[(ka + 1) * ldb];
    // 8 args: (neg_a, A, neg_b, B, c_mod, C, reuse_a, reuse_b)
    c = __builtin_amdgcn_wmma_f32_16x16x4_f32(false, a, false, b,
                                              (short)0, c, false, false);
  }
  return c;
}

// ---------------------------------------------------------------------------
// Kernel 2: fused MLP + softmax for a 16-row batch tile per block.
// 512 threads = 16 wave32s; layer1 = 16 WMMA column tiles, layer2 = 4.
// Activation tile staged memory->LDS with GLOBAL_LOAD_ASYNC_TO_LDS_B128
// (ASYNCcnt path), drained with s_wait_asynccnt before the barrier.
// ---------------------------------------------------------------------------
__global__ void __launch_bounds__(512)
mlp_kernel(const float* __restrict__ vectors,
           const float* __restrict__ W1, const float* __restrict__ b1,
           const float* __restrict__ W2, const float* __restrict__ b2,
           const float* __restrict__ W3, const float* __restrict__ b3,
           float* __restrict__ out) {
  // Padded strides (stride % 64 == 4) so the 16 A-rows hit distinct LDS banks.
  __shared__ float vec_s[16][DSZ + 4];
  __shared__ float h1_s[16][H1SZ + 4];
  __shared__ float h2_s[16][H2SZ + 4];

  const int tid   = threadIdx.x;
  const int lane  = tid & 31;
  const int wave  = tid >> 5;
  const int tileB = blockIdx.x;  // 64 tiles of 16 batch rows

  // Stage the 16x768 activation tile into LDS: 3072 float4 over 512 threads
  // = exactly 6 async B128 copies per lane (no remainder -> EXEC all-ones).
  {
    const float4* __restrict__ src =
        (const float4*)(vectors + (size_t)tileB * 16 * DSZ);
#pragma unroll
    for (int i = tid; i < 16 * (DSZ / 4); i += 512) {
      const int r  = i / (DSZ / 4);
      const int c4 = i % (DSZ / 4);
      const uint32_t lds_off =
          (uint32_t)(uintptr_t)&vec_s[r][c4 * 4];        // LDS byte offset
      const float4* gaddr = src + r * (DSZ / 4) + c4;
      asm volatile("global_load_async_to_lds_b128 %0, %1, off"
                   :
                   : "v"(lds_off), "v"(gaddr)
                   : "memory");
    }
    asm volatile("s_wait_asynccnt 0" ::: "memory");
  }
  __syncthreads();

  // Layer 1: [16,768] @ [768,256] + b1, relu.  One 16-col tile per wave.
  {
    const int n0  = wave * 16;
    const int col = n0 + (lane & 15);
    v8f c = wmma16x16_f32(&vec_s[0][0], DSZ + 4, W1 + col, H1SZ, DSZ, lane);
    const float bias = b1[col];
    const int mb = (lane >> 4) << 3;  // D layout: VGPR r = rows r / r+8
#pragma unroll
    for (int r = 0; r < 8; ++r) {
      const float v = c[r] + bias;
      h1_s[mb + r][col] = v > 0.f ? v : 0.f;
    }
  }
  __syncthreads();

  // Layer 2: [16,256] @ [256,64] + b2, relu.  Waves 0-3 (wave-uniform branch,
  // EXEC stays all-ones inside the WMMA loop).
  if (wave < 4) {
    const int n0  = wave * 16;
    const int col = n0 + (lane & 15);
    v8f c = wmma16x16_f32(&h1_s[0][0], H1SZ + 4, W2 + col, H2SZ, H1SZ, lane);
    const float bias = b2[col];
    const int mb = (lane >> 4) << 3;
#pragma unroll
    for (int r = 0; r < 8; ++r) {
      const float v = c[r] + bias;
      h2_s[mb + r][col] = v > 0.f ? v : 0.f;
    }
  }
  __syncthreads();

  // Layer 3 + softmax: 16 rows x 2 logits, trivial scalar work.
  if (tid < 16) {
    float l0 = b3[0], l1 = b3[1];
#pragma unroll 8
    for (int k = 0; k < H2SZ; ++k) {
      const float h = h2_s[tid][k];
      l0 = fmaf(h, W3[k * 2 + 0], l0);
      l1 = fmaf(h, W3[k * 2 + 1], l1);
    }
    const float m  = fmaxf(l0, l1);
    const float e0 = __expf(l0 - m);
    const float e1 = __expf(l1 - m);
    const float inv = 1.f / (e0 + e1);
    const int rowg = tileB * 16 + tid;
    out[rowg * 2 + 0] = e0 * inv;
    out[rowg * 2 + 1] = e1 * inv;
  }
}

// ---------------------------------------------------------------------------
extern "C" void kernel_launch(void* const* d_in, const int* in_sizes, int n_in,
                              void* d_out, int out_size, void* d_ws,
                              size_t ws_size, hipStream_t stream) {
  (void)in_sizes; (void)n_in; (void)out_size; (void)ws_size;
  const float*   embeds = (const float*)d_in[0];
  const uint8_t* mask   = (const uint8_t*)d_in[1];  // jnp bool -> 1 byte
  const float*   W1     = (const float*)d_in[2];
  const float*   b1     = (const float*)d_in[3];
  const float*   W2     = (const float*)d_in[4];
  const float*   b2     = (const float*)d_in[5];
  const float*   W3     = (const float*)d_in[6];
  const float*   b3     = (const float*)d_in[7];
  float* out = (float*)d_out;

  float* wA      = (float*)d_ws;   // 127 weights (padded to 128)
  float* vectors = wA + 128;       // [1024, 768] fp32 ~ 3MB

  spec_weights_kernel<<<1, 128, 0, stream>>>(wA);
  spec_reduce_kernel<<<BSZ, 192, 0, stream>>>(embeds, mask, wA, vectors);
  mlp_kernel<<<BSZ / 16, 512, 0, stream>>>(vectors, W1, b1, W2, b2, W3, b3,
                                           out);
}